// SymmetryControl2_48034914238617
// MI455X (gfx1250) — compile-verified
//
#include <hip/hip_runtime.h>
#include <stdint.h>

// Problem shape (fixed by the reference): B=8, C=32, H=256, W=256, step=32.
#define B_   8
#define C_   32
#define H_   256
#define W_   256
#define HW_  (H_ * W_)   // 65536 elements per (b,c) plane
#define CPG  8           // channels per wave-task
#define GRPS (C_ / CPG)  // 4 wave-tasks per (b,h) row

// Row layout per wave (wave32): lane l, register m hold position 32*m + l.
// roll by 32k  -> same lane, register (m-k)&7          (pure register re-index)
// flip about 32(j+1)-1 -> lane 31-l, register (j-m)&7  (one lane-reversal shuffle)

__device__ __forceinline__ void load_row_nt(float (&r)[8], const float* p) {
#pragma unroll
  for (int m = 0; m < 8; ++m)
    r[m] = __builtin_nontemporal_load(p + 32 * m);   // streamed: read exactly once
}

__device__ __forceinline__ void process_row(const float (&xr)[8],
                                            const float (&wv)[8],
                                            const float (&rden)[8],
                                            const float (&g)[11],
                                            float* op) {
  float wx[8];
#pragma unroll
  for (int m = 0; m < 8; ++m) wx[m] = wv[m] * xr[m];
  float rwx[8];                                      // wx with lanes reversed
#pragma unroll
  for (int m = 0; m < 8; ++m) rwx[m] = __shfl_xor(wx[m], 31, 32);
#pragma unroll
  for (int m = 0; m < 8; ++m) {
    float acc = wx[m];                               // identity term
#pragma unroll
    for (int k = 1; k <= 7; ++k)                     // 7 circular shifts
      acc = fmaf(g[k - 1], wx[(m - k) & 7], acc);
#pragma unroll
    for (int j = 0; j < 4; ++j)                      // 4 reflections
      acc = fmaf(g[7 + j], rwx[(j - m) & 7], acc);
    __builtin_nontemporal_store(acc * rden[m], op + 32 * m);
  }
}

__global__ __launch_bounds__(256) void SymmetryControl2_kernel(
    const float* __restrict__ x,   // [B,C,H,W]
    const float* __restrict__ s,   // [B,11]
    const float* __restrict__ w,   // [B,1,H,W]
    const float* __restrict__ wp,  // [1]
    const float* __restrict__ bp,  // [1]
    float* __restrict__ out)       // [B,C,H,W]
{
  __shared__ float sh_sg[11];

  const int tid  = threadIdx.x;
  const int l    = tid & 31;        // lane
  const int wvid = tid >> 5;        // wave in block (0..7)
  // task = (row, channel-group): 8 wave-tasks per block = 2 rows x 4 groups.
  const int row  = blockIdx.x * 2 + (wvid >> 2);   // 0..B*H-1
  const int grp  = wvid & 3;                       // channel group, 0..3
  const int b    = row >> 8;
  const int h    = row & (H_ - 1);

  // Gates (uniform per batch; both rows of this block share b).
  if (tid < 11) {
    const int bb = blockIdx.x >> 7;                 // block-uniform batch index
    float v = wp[0] * s[bb * 11 + tid] + bp[0];
    sh_sg[tid] = 1.0f / (1.0f + expf(-v));
  }
  __syncthreads();                                  // only barrier in the kernel

  float g[11];
#pragma unroll
  for (int t = 0; t < 11; ++t) g[t] = sh_sg[t];

  // w row in strided registers + reversed copy; denominator & reciprocal once.
  const float* wrow = w + ((size_t)b * H_ + h) * W_ + l;
  float wv[8];
  load_row_nt(wv, wrow);
  float rwv[8];
#pragma unroll
  for (int m = 0; m < 8; ++m) rwv[m] = __shfl_xor(wv[m], 31, 32);
  float rden[8];
#pragma unroll
  for (int m = 0; m < 8; ++m) {
    float d = wv[m];
#pragma unroll
    for (int k = 1; k <= 7; ++k) d = fmaf(g[k - 1], wv[(m - k) & 7], d);
#pragma unroll
    for (int j = 0; j < 4; ++j)  d = fmaf(g[7 + j], rwv[(j - m) & 7], d);
    rden[m] = 1.0f / d;
  }

  // This wave's 8 channels, register double-buffered.
  const size_t plane0 = (size_t)b * C_ * HW_ + (size_t)(grp * CPG) * HW_ +
                        (size_t)h * W_ + l;
  const float* xrow = x + plane0;
  float*       orow = out + plane0;

  float xa[8], xb[8];
  load_row_nt(xa, xrow);                            // channel 0 of this group
#pragma unroll
  for (int c = 0; c < CPG; c += 2) {
    load_row_nt(xb, xrow + (size_t)(c + 1) * HW_);  // prefetch next channel
    process_row(xa, wv, rden, g, orow + (size_t)c * HW_);
    if (c + 2 < CPG)
      load_row_nt(xa, xrow + (size_t)(c + 2) * HW_);
    process_row(xb, wv, rden, g, orow + (size_t)(c + 1) * HW_);
  }
}

extern "C" void kernel_launch(void* const* d_in, const int* in_sizes, int n_in,
                              void* d_out, int out_size, void* d_ws, size_t ws_size,
                              hipStream_t stream) {
  (void)in_sizes; (void)n_in; (void)d_ws; (void)ws_size; (void)out_size;
  const float* x  = (const float*)d_in[0];
  const float* s  = (const float*)d_in[1];
  const float* w  = (const float*)d_in[2];
  const float* wp = (const float*)d_in[3];
  const float* bp = (const float*)d_in[4];
  float* out = (float*)d_out;

  // 8192 wave-tasks = 2048 rows x 4 channel-groups; 8 waves per block.
  dim3 grid((B_ * H_ * GRPS) / 8);   // 1024 blocks
  dim3 block(256);                   // 8 wave32s
  SymmetryControl2_kernel<<<grid, block, 0, stream>>>(x, s, w, wp, bp, out);
}